// CayleyMengerValidator_57071525430055
// MI455X (gfx1250) — compile-verified
//
#include <hip/hip_runtime.h>
#include <math.h>

typedef __attribute__((ext_vector_type(2))) float v2f;
typedef __attribute__((ext_vector_type(8))) float v8f;

#define B_DIM       16
#define O_DIM       8192
#define NPTS        5
#define D_DIM       64
#define WAVES_PB    4
#define BLOCK_THR   (WAVES_PB * 32)
#define SIMP_PW     3                    // 3 simplices * 5 pts = 15 of 16 WMMA rows
#define SIMP_PB     (WAVES_PB * SIMP_PW)
#define ROW_STRIDE  68                   // 64 + 4 pad: 272B rows -> 16B aligned, bank-spread
#define G_STRIDE    17

__global__ __launch_bounds__(BLOCK_THR)
void cm_main(const float* __restrict__ pred, const int* __restrict__ sidx,
             float* __restrict__ partials, int n_samples, int S)
{
    const int lane = threadIdx.x & 31;
    const int wave = threadIdx.x >> 5;
    const int half = lane >> 4;
    const int lrow = lane & 15;

    __shared__ float ldsV[WAVES_PB][NPTS * SIMP_PW * ROW_STRIDE];
    __shared__ float ldsG[WAVES_PB][16 * G_STRIDE];
    __shared__ float ldsP[SIMP_PB];

    const int s_base = (blockIdx.x * WAVES_PB + wave) * SIMP_PW;

    // ---- gather: 3 simplices x (5x64 f32) -> LDS, float4-coalesced ----
    for (int si = 0; si < SIMP_PW; ++si) {
        int s  = s_base + si;
        int sc = s < S ? s : S - 1;                       // clamp; masked after WMMA
        int b  = sc / n_samples;
        int j  = sc - b * n_samples;
        int idx = sidx[b * n_samples + j];
        idx = idx < 0 ? 0 : (idx >= O_DIM ? O_DIM - 1 : idx);
        const float4* src =
            (const float4*)(pred + (size_t)(b * O_DIM + idx) * (NPTS * D_DIM));
        for (int t = lane; t < NPTS * (D_DIM / 4); t += 32) {
            int row = t >> 4;                             // 16 float4 per 64-f32 row
            int q   = t & 15;
            ((float4*)&ldsV[wave][(si * NPTS + row) * ROW_STRIDE])[q] = src[t];
        }
    }
    __syncthreads();

    // ---- Gram G = V * V^T via V_WMMA_F32_16X16X4_F32, K=64 as 16 chained steps.
    // A (16x4 f32): lanes 0-15 row M=lane {K0,K1}; lanes 16-31 same rows {K2,K3}.
    // B (4x16) of V^T has the identical register image -> reuse fragment as A and B.
    const int arow = (lrow < NPTS * SIMP_PW) ? lrow : (NPTS * SIMP_PW - 1);
    const float* vr = &ldsV[wave][arow * ROW_STRIDE];
    v8f acc = {};
#pragma unroll
    for (int kk = 0; kk < 16; ++kk) {
        v2f a;
        a.x = vr[kk * 4 + 2 * half + 0];
        a.y = vr[kk * 4 + 2 * half + 1];
        acc = __builtin_amdgcn_wmma_f32_16x16x4_f32(
            false, a, false, a, (short)0, acc, false, false);
    }
    // D layout: VGPR v, lane (h,n) -> G[h*8+v][n]
#pragma unroll
    for (int v = 0; v < 8; ++v)
        ldsG[wave][(half * 8 + v) * G_STRIDE + lrow] = acc[v];
    __syncthreads();

    // ---- per-simplex scalar epilogue (lanes 0..2 of each wave) ----
    float contrib = 0.f;
    if (lane < SIMP_PW) {
        const int s = s_base + lane;
        if (s < S) {
            const float* Gp = &ldsG[wave][lane * NPTS * G_STRIDE + lane * NPTS];
            float g[NPTS][NPTS];
#pragma unroll
            for (int i = 0; i < NPTS; ++i)
#pragma unroll
                for (int j = 0; j < NPTS; ++j)
                    g[i][j] = Gp[i * G_STRIDE + j];

            // rose score from centered Gram: r_ij = g_ij - m_i - m_j + mm
            float m[NPTS], mm = 0.f;
#pragma unroll
            for (int i = 0; i < NPTS; ++i) {
                float ssum = 0.f;
#pragma unroll
                for (int j = 0; j < NPTS; ++j) ssum += g[i][j];
                m[i] = ssum * 0.2f;
                mm += m[i];
            }
            mm *= 0.2f;
            float nrm[NPTS];
#pragma unroll
            for (int i = 0; i < NPTS; ++i) {
                float rii = g[i][i] - 2.f * m[i] + mm;
                nrm[i] = fmaxf(sqrtf(fmaxf(rii, 0.f)), 1e-8f);
            }
            float cs = 0.f;
#pragma unroll
            for (int i = 0; i < NPTS; ++i)
#pragma unroll
                for (int j = 0; j < NPTS; ++j)
                    if (j > i) {
                        float rij = g[i][j] - m[i] - m[j] + mm;
                        cs += 2.f * (rij / (nrm[i] * nrm[j]));
                    }
            float rose = (cs * (1.f / 20.f) + 1.f) * 0.5f;

            // pairwise sq-dists, regularity
            float d2[NPTS][NPTS];
            float dmin = 3.4e38f, dmax = 0.f;
#pragma unroll
            for (int i = 0; i < NPTS; ++i) d2[i][i] = 0.f;
#pragma unroll
            for (int i = 0; i < NPTS; ++i)
#pragma unroll
                for (int j = 0; j < NPTS; ++j)
                    if (j > i) {
                        float v = g[i][i] + g[j][j] - 2.f * g[i][j];
                        d2[i][j] = v; d2[j][i] = v;
                        float d = sqrtf(fmaxf(v, 1e-12f));
                        dmin = fminf(dmin, d);
                        dmax = fmaxf(dmax, d);
                    }
            float reg = dmin / fmaxf(dmax, 1e-8f);

            // Cayley-Menger 6x6 determinant (partial-pivot LU)
            float cm[6][6];
            cm[0][0] = 0.f;
            for (int j = 1; j < 6; ++j) cm[0][j] = 1.f;
            for (int i = 1; i < 6; ++i) {
                cm[i][0] = 1.f;
                for (int j = 1; j < 6; ++j) cm[i][j] = d2[i - 1][j - 1];
            }
            float det = 1.f;
            for (int c = 0; c < 6; ++c) {
                int p = c; float best = fabsf(cm[c][c]);
                for (int r2 = c + 1; r2 < 6; ++r2) {
                    float a = fabsf(cm[r2][c]);
                    if (a > best) { best = a; p = r2; }
                }
                if (p != c) {
                    for (int cc = 0; cc < 6; ++cc) {
                        float t = cm[c][cc]; cm[c][cc] = cm[p][cc]; cm[p][cc] = t;
                    }
                    det = -det;
                }
                float piv = cm[c][c];
                det *= piv;
                if (piv != 0.f) {
                    float inv = 1.f / piv;
                    for (int r2 = c + 1; r2 < 6; ++r2) {
                        float f = cm[r2][c] * inv;
                        for (int cc = c + 1; cc < 6; ++cc)
                            cm[r2][cc] = fmaf(-f, cm[c][cc], cm[r2][cc]);
                    }
                }
            }
            float v2 = det * (-1.f / 9216.f);            // (-1)^5 / (2^4 * (4!)^2)
            float vol = sqrtf(fmaxf(v2, 0.f));
            float degen = (vol < 1e-8f) ? 1.f : 0.f;

            contrib = 0.5f * (1.f - rose) + 0.3f * (1.f - reg) + 0.2f * degen;
        }
        ldsP[wave * SIMP_PW + lane] = contrib;
    }
    __syncthreads();

    if (threadIdx.x == 0) {
        float sum = 0.f;
        for (int i = 0; i < SIMP_PB; ++i) sum += ldsP[i];   // fixed order: deterministic
        partials[blockIdx.x] = sum;
    }
}

__global__ void cm_finalize(const float* __restrict__ partials,
                            float* __restrict__ out, int n, float invS)
{
    if (threadIdx.x == 0 && blockIdx.x == 0) {
        float s = 0.f;
        for (int i = 0; i < n; ++i) s += partials[i];       // fixed order: deterministic
        out[0] = s * invS;
    }
}

extern "C" void kernel_launch(void* const* d_in, const int* in_sizes, int n_in,
                              void* d_out, int out_size, void* d_ws, size_t ws_size,
                              hipStream_t stream)
{
    const float* pred = (const float*)d_in[0];
    const int*   sidx = (const int*)d_in[1];

    const int S         = in_sizes[1];        // B * n_samples = total simplices
    const int n_samples = S / B_DIM;
    const int nblocks   = (S + SIMP_PB - 1) / SIMP_PB;

    float* partials = (float*)d_ws;           // nblocks floats, fully overwritten

    cm_main<<<nblocks, BLOCK_THR, 0, stream>>>(pred, sidx, partials, n_samples, S);
    cm_finalize<<<1, 32, 0, stream>>>(partials, (float*)d_out, nblocks,
                                      1.0f / (float)S);
}